// Histogram_15126874816754
// MI455X (gfx1250) — compile-verified
//
#include <hip/hip_runtime.h>

// ---------------------------------------------------------------------------
// Event-camera histogram: scatter-add 8M events into [B, 2, H, W] then clip.
//
// Roofline (MI455X, 23.3 TB/s HBM, 192 MB L2):
//   events  : 8M x 16 B SoA   = 128 MB, streamed once  -> non-temporal loads
//   output  : 8*2*720*1280*4B =  59 MB                 -> fits in L2; atomics
//             resolve in L2 (global_atomic_add_f32, no-return form)
//   passes  : zero (59 MB W) + scatter (128 MB R) + clip (59 R + 59 W)
//             ~305 MB -> ~13-15 us floor. NT hints keep the accumulator
//             L2-resident while the event stream passes through.
// No matrix structure in this workload -> WMMA inapplicable by design.
// ---------------------------------------------------------------------------

typedef int   __attribute__((ext_vector_type(4))) v4i;
typedef float __attribute__((ext_vector_type(4))) v4f;

// Pass 1: zero the accumulator with regular-temporal b128 stores so the
// lines are dirty + resident in L2 before the atomic storm.
__global__ void hist_zero_kernel(float* __restrict__ out, int n4) {
    int i = blockIdx.x * blockDim.x + threadIdx.x;
    if (i < n4) {
        v4f z = {0.0f, 0.0f, 0.0f, 0.0f};
        reinterpret_cast<v4f*>(out)[i] = z;
    }
}

__global__ void hist_zero_tail_kernel(float* __restrict__ out, int base, int n) {
    int i = base + blockIdx.x * blockDim.x + threadIdx.x;
    if (i < n) out[i] = 0.0f;
}

// Pass 2: scatter. 8 events per lane via 2x 128-bit coalesced non-temporal
// loads per SoA stream (8 outstanding b128 loads/lane to hide HBM latency),
// then 8 global_atomic_add_f32 (no-return form, STOREcnt; never waited on
// per-op).
__global__ void hist_scatter_kernel(const int*   __restrict__ xs,
                                    const int*   __restrict__ ys,
                                    const float* __restrict__ ps,
                                    const int*   __restrict__ bs,
                                    const int*   __restrict__ wptr,
                                    const int*   __restrict__ hptr,
                                    float*       __restrict__ out,
                                    int n8) {
    int i = blockIdx.x * blockDim.x + threadIdx.x;
    if (i >= n8) return;

    const int W = *wptr;   // uniform -> scalar load path
    const int H = *hptr;

    const v4i* x4 = reinterpret_cast<const v4i*>(xs);
    const v4i* y4 = reinterpret_cast<const v4i*>(ys);
    const v4f* p4 = reinterpret_cast<const v4f*>(ps);
    const v4i* b4 = reinterpret_cast<const v4i*>(bs);

    int j = i << 1;  // two v4 groups per lane

    // Issue all 8 NT loads up front (compiler clauses these) before any use.
    v4i xv0 = __builtin_nontemporal_load(x4 + j);
    v4i xv1 = __builtin_nontemporal_load(x4 + j + 1);
    v4i yv0 = __builtin_nontemporal_load(y4 + j);
    v4i yv1 = __builtin_nontemporal_load(y4 + j + 1);
    v4f pv0 = __builtin_nontemporal_load(p4 + j);
    v4f pv1 = __builtin_nontemporal_load(p4 + j + 1);
    v4i bv0 = __builtin_nontemporal_load(b4 + j);
    v4i bv1 = __builtin_nontemporal_load(b4 + j + 1);

#pragma unroll
    for (int k = 0; k < 4; ++k) {
        float p   = pv0[k];
        int   pos = (p > 0.0f) ? 1 : 0;                    // polarity bin
        int   idx = (((bv0[k] << 1) + pos) * H + yv0[k]) * W + xv0[k];
        atomicAdd(out + idx, __builtin_fabsf(p) * 0.05f);  // (|p| - 0) / 20
    }
#pragma unroll
    for (int k = 0; k < 4; ++k) {
        float p   = pv1[k];
        int   pos = (p > 0.0f) ? 1 : 0;
        int   idx = (((bv1[k] << 1) + pos) * H + yv1[k]) * W + xv1[k];
        atomicAdd(out + idx, __builtin_fabsf(p) * 0.05f);
    }
}

// Scalar tail (only launched if N % 8 != 0 — not the case for N = 8M).
__global__ void hist_scatter_tail_kernel(const int*   __restrict__ xs,
                                         const int*   __restrict__ ys,
                                         const float* __restrict__ ps,
                                         const int*   __restrict__ bs,
                                         const int*   __restrict__ wptr,
                                         const int*   __restrict__ hptr,
                                         float*       __restrict__ out,
                                         int base, int n) {
    int i = base + blockIdx.x * blockDim.x + threadIdx.x;
    if (i >= n) return;
    const int W = *wptr;
    const int H = *hptr;
    float p   = ps[i];
    int   pos = (p > 0.0f) ? 1 : 0;
    int   idx = (((bs[i] << 1) + pos) * H + ys[i]) * W + xs[i];
    atomicAdd(out + idx, __builtin_fabsf(p) * 0.05f);
}

// Pass 3: clip to 1.0. Single-touch read/write -> non-temporal both ways
// (drains the accumulator out of L2 as we finish).
__global__ void hist_clip_kernel(float* __restrict__ out, int n4) {
    int i = blockIdx.x * blockDim.x + threadIdx.x;
    if (i < n4) {
        v4f v = __builtin_nontemporal_load(reinterpret_cast<const v4f*>(out) + i);
        v4f r;
#pragma unroll
        for (int k = 0; k < 4; ++k) r[k] = __builtin_fminf(v[k], 1.0f);
        __builtin_nontemporal_store(r, reinterpret_cast<v4f*>(out) + i);
    }
}

__global__ void hist_clip_tail_kernel(float* __restrict__ out, int base, int n) {
    int i = base + blockIdx.x * blockDim.x + threadIdx.x;
    if (i < n) out[i] = __builtin_fminf(out[i], 1.0f);
}

extern "C" void kernel_launch(void* const* d_in, const int* in_sizes, int n_in,
                              void* d_out, int out_size, void* d_ws, size_t ws_size,
                              hipStream_t stream) {
    const int*   xs   = (const int*)  d_in[0];
    const int*   ys   = (const int*)  d_in[1];
    const float* ps   = (const float*)d_in[2];
    const int*   bs   = (const int*)  d_in[3];
    const int*   wptr = (const int*)  d_in[4];  // width  (device scalar)
    const int*   hptr = (const int*)  d_in[5];  // height (device scalar)
    // d_in[6] = batch_size, implicit in out_size

    float* out = (float*)d_out;
    const int N = in_sizes[0];

    const int BLOCK = 256;  // 8 wave32 waves per workgroup

    // ---- Pass 1: zero accumulator ----
    int out4     = out_size >> 2;
    int out_tail = out_size - (out4 << 2);
    if (out4 > 0)
        hipLaunchKernelGGL(hist_zero_kernel,
                           dim3((out4 + BLOCK - 1) / BLOCK), dim3(BLOCK), 0, stream,
                           out, out4);
    if (out_tail > 0)
        hipLaunchKernelGGL(hist_zero_tail_kernel,
                           dim3(1), dim3(BLOCK), 0, stream,
                           out, out4 << 2, out_size);

    // ---- Pass 2: scatter-add ----
    int n8   = N >> 3;
    int tail = N - (n8 << 3);
    if (n8 > 0)
        hipLaunchKernelGGL(hist_scatter_kernel,
                           dim3((n8 + BLOCK - 1) / BLOCK), dim3(BLOCK), 0, stream,
                           xs, ys, ps, bs, wptr, hptr, out, n8);
    if (tail > 0)
        hipLaunchKernelGGL(hist_scatter_tail_kernel,
                           dim3(1), dim3(BLOCK), 0, stream,
                           xs, ys, ps, bs, wptr, hptr, out, n8 << 3, N);

    // ---- Pass 3: clip ----
    if (out4 > 0)
        hipLaunchKernelGGL(hist_clip_kernel,
                           dim3((out4 + BLOCK - 1) / BLOCK), dim3(BLOCK), 0, stream,
                           out, out4);
    if (out_tail > 0)
        hipLaunchKernelGGL(hist_clip_tail_kernel,
                           dim3(1), dim3(BLOCK), 0, stream,
                           out, out4 << 2, out_size);
}